// MultiHeadAttentionQuantum_65481071406393
// MI455X (gfx1250) — compile-verified
//
#include <hip/hip_runtime.h>

// MI455X / gfx1250, wave32.
// out[b,s,f] = sum_e z[e] * W[f,e] + bias[f], identical for every (b,s).
// Kernel 1: compute z (quantum circuit, input-independent) + y = W z + b via V_WMMA_F32_16X16X4_F32.
// Kernel 2: broadcast y over 65536 rows with non-temporal b128 stores (HBM-bound, ~256 MiB).

typedef __attribute__((ext_vector_type(2))) float v2f;
typedef __attribute__((ext_vector_type(4))) float v4f;
typedef __attribute__((ext_vector_type(8))) float v8f;

#define EDIM 1024
#define NROWS 65536              // B*S = 8*8192
#define BCAST_BLOCKS 2048
#define ROWS_PER_BLOCK (NROWS / BCAST_BLOCKS)   // 32

// ---------------------------------------------------------------------------
// Kernel 1: 8 blocks x 256 threads. Each block:
//   - cooperatively computes z[0..1023] into LDS,
//   - 8 waves each produce a 16-row tile of y via wmma_f32_16x16x4_f32.
// ---------------------------------------------------------------------------
__global__ __launch_bounds__(256) void qz_matvec_wmma_kernel(
    const float* __restrict__ bs_theta, const float* __restrict__ bs_phi,
    const float* __restrict__ phases, const float* __restrict__ squeeze_r,
    const float* __restrict__ disp_r, const float* __restrict__ kerr,
    const float* __restrict__ W, const float* __restrict__ bcomb,
    float* __restrict__ y) {
  __shared__ float zsh[EDIM];
  const int t = threadIdx.x;

  // ---- per-wire qubit simulation (complex 2-amplitude state) ----
  const float th0 = bs_theta[0];
  const float th1 = bs_phi[0];
  const float c0 = cosf(0.5f * th0), s0 = sinf(0.5f * th0);
  const float c1 = cosf(0.5f * th1), s1 = sinf(0.5f * th1);

  for (int j = 0; j < EDIM / 256; ++j) {
    const int e = t + 256 * j;
    // |0> --rx(bs_theta)-->  a = (c0, 0), b = (0, -s0)
    float ar = c0, ai = 0.0f, br = 0.0f, bi = -s0;
    // ry(bs_phi): a' = c a - s b ; b' = s a + c b   (real coefficients)
    {
      float nar = c1 * ar - s1 * br, nai = c1 * ai - s1 * bi;
      float nbr = s1 * ar + c1 * br, nbi = s1 * ai + c1 * bi;
      ar = nar; ai = nai; br = nbr; bi = nbi;
    }
    // rz(phases[e]): a *= exp(-i h), b *= exp(+i h), h = phase/2
    {
      float h = 0.5f * phases[e];
      float cp = cosf(h), sp = sinf(h);
      float nar = cp * ar + sp * ai, nai = cp * ai - sp * ar;
      float nbr = cp * br - sp * bi, nbi = cp * bi + sp * br;
      ar = nar; ai = nai; br = nbr; bi = nbi;
    }
    // ry(squeeze_r[e])
    {
      float h = 0.5f * squeeze_r[e];
      float c = cosf(h), s = sinf(h);
      float nar = c * ar - s * br, nai = c * ai - s * bi;
      float nbr = s * ar + c * br, nbi = s * ai + c * bi;
      ar = nar; ai = nai; br = nbr; bi = nbi;
    }
    // rx(displacement_r[e]): a' = c a - i s b ; b' = -i s a + c b
    {
      float h = 0.5f * disp_r[e];
      float c = cosf(h), s = sinf(h);
      float nar = c * ar + s * bi, nai = c * ai - s * br;
      float nbr = s * ai + c * br, nbi = -s * ar + c * bi;
      ar = nar; ai = nai; br = nbr; bi = nbi;
    }
    // rz(kerr[e]) multiplies a,b by unit-magnitude phases only:
    // |a|^2 and |b|^2 are invariant, so <Z> is unchanged -> skip exactly.
    zsh[e] = (ar * ar + ai * ai) - (br * br + bi * bi);
  }
  (void)kerr;
  __syncthreads();

  // ---- y-tile = W[ftile:ftile+16, :] * z  via WMMA f32 16x16x4 ----
  const int wave = t >> 5;       // 0..7
  const int lane = t & 31;
  const int hi   = lane >> 4;    // 0: K=0,1 ; 1: K=2,3  (32-bit A/B layout)
  const int m    = lane & 15;    // A-matrix row M
  const int ftile = blockIdx.x * 128 + wave * 16;
  const float* __restrict__ Wrow = W + (size_t)(ftile + m) * EDIM;

  v8f acc = {};
  for (int kb = 0; kb < EDIM; kb += 4) {
    const int k = kb + 2 * hi;
    v2f a, b;
    a.x = Wrow[k];
    a.y = Wrow[k + 1];
    // B: z broadcast into all 16 columns -> every column of D equals y-tile.
    b.x = zsh[k];
    b.y = zsh[k + 1];
    acc = __builtin_amdgcn_wmma_f32_16x16x4_f32(
        /*neg_a=*/false, a, /*neg_b=*/false, b,
        /*c_mod=*/(short)0, acc, /*reuse_a=*/false, /*reuse_b=*/false);
  }

  // D layout (16x16 f32): VGPR r -> M = r + 8*hi for lane group; N = lane&15.
  // Take column N=0: lane 0 holds M=0..7, lane 16 holds M=8..15.
  if (m == 0) {
    #pragma unroll
    for (int r = 0; r < 8; ++r) {
      const int f = ftile + 8 * hi + r;
      y[f] = acc[r] + bcomb[f];
    }
  }
}

// ---------------------------------------------------------------------------
// Kernel 2: broadcast y (1024 f32 = 256 float4) over 65536 rows.
// Each thread registers its float4 of y, then streams ROWS_PER_BLOCK
// non-temporal b128 stores (output > L2, never re-read on device).
// ---------------------------------------------------------------------------
__global__ __launch_bounds__(256) void broadcast_rows_kernel(
    const float* __restrict__ y, float* __restrict__ out) {
  const int t = threadIdx.x;                       // float4 column 0..255
  const v4f v = ((const v4f*)y)[t];
  v4f* __restrict__ out4 = (v4f*)out;
  const size_t row0 = (size_t)blockIdx.x * ROWS_PER_BLOCK;
  #pragma unroll 4
  for (int r = 0; r < ROWS_PER_BLOCK; ++r) {
    __builtin_nontemporal_store(v, &out4[(row0 + r) * 256 + t]);
  }
}

extern "C" void kernel_launch(void* const* d_in, const int* in_sizes, int n_in,
                              void* d_out, int out_size, void* d_ws, size_t ws_size,
                              hipStream_t stream) {
  // setup_inputs() order: x, bs_theta, bs_phi, phases, squeeze_r,
  //                       displacement_r, kerr, W_combine, b_combine
  const float* bs_theta = (const float*)d_in[1];
  const float* bs_phi   = (const float*)d_in[2];
  const float* phases   = (const float*)d_in[3];
  const float* squeezer = (const float*)d_in[4];
  const float* disp_r   = (const float*)d_in[5];
  const float* kerr     = (const float*)d_in[6];
  const float* W        = (const float*)d_in[7];
  const float* bcomb    = (const float*)d_in[8];
  float* y   = (float*)d_ws;       // 1024 floats of scratch
  float* out = (float*)d_out;      // 65536 x 1024 f32

  qz_matvec_wmma_kernel<<<EDIM / 128, 256, 0, stream>>>(
      bs_theta, bs_phi, phases, squeezer, disp_r, kerr, W, bcomb, y);
  broadcast_rows_kernel<<<BCAST_BLOCKS, 256, 0, stream>>>(y, out);
}